// DistributedRealSHT_63024350101995
// MI455X (gfx1250) — compile-verified
//
#include <hip/hip_runtime.h>
#include <hip/hip_bf16.h>

// ---------------------------------------------------------------------------
// Distributed real SHT for MI455X (gfx1250), fp32 WMMA pipeline.
//   Stage 0: build DFT basis  Ft[m2][n]  (m2<361: cos, 361..721: -sin), *2pi/720
//   Stage 1: Xf[m2][row] = sum_n X[row][n] * Ft[m2][n]   (NT GEMM, WMMA f32,
//            async global->LDS double-buffered)
//   Stage 2: out[c][l][m][ri] = sum_k Xf[plane][c*361+k] * W[m][l][k]
// ---------------------------------------------------------------------------

#define SHT_NLAT 361
#define SHT_NLON 720
#define SHT_MMAX 361
#define SHT_LMAX 360
#define SHT_NC   256
#define SHT_M1   (SHT_NC * SHT_NLAT)   // 92416 rows, = 722 * 128 exactly
#define SHT_N1   (2 * SHT_MMAX)        // 722 planes (re then im)
#define SHT_N1P  768                   // padded planes so B-tile loads need no guard
#define SHT_K1   SHT_NLON              // 720

#define LDSROW   20                    // padded row stride (floats): 16B-aligned rows
#define BLK      128                   // block tile M=N=128
#define BLKK     16                    // K tile

typedef __attribute__((ext_vector_type(2))) float v2f;
typedef __attribute__((ext_vector_type(4))) float v4f;
typedef __attribute__((ext_vector_type(8))) float v8f;
typedef __attribute__((ext_vector_type(4))) int   v4i;

#if defined(__has_builtin)
#if __has_builtin(__builtin_amdgcn_global_load_async_to_lds_b128) && \
    __has_builtin(__builtin_amdgcn_s_wait_asynccnt)
#define SHT_HAVE_ASYNC_LDS 1
#endif
#endif

#ifdef SHT_HAVE_ASYNC_LDS
typedef __attribute__((address_space(1))) v4i gv4i;   // global int4
typedef __attribute__((address_space(3))) v4i lv4i;   // LDS int4
__device__ __forceinline__ void async_b128(const float* g, float* l) {
    float* gnc = const_cast<float*>(g);
    __builtin_amdgcn_global_load_async_to_lds_b128((gv4i*)gnc, (lv4i*)l, 0, 0);
}
#endif

// ---------------------------------------------------------------------------
// Stage 0: DFT basis. Exact integer phase reduction keeps fp32 trig accurate.
// Fills padded 768 rows (722 valid, rest zero).
// ---------------------------------------------------------------------------
__global__ void sht_build_basis(float* __restrict__ Ft) {
    long idx = (long)blockIdx.x * blockDim.x + threadIdx.x;
    long total = (long)SHT_N1P * SHT_K1;
    if (idx >= total) return;
    int n  = (int)(idx % SHT_K1);
    int m2 = (int)(idx / SHT_K1);
    float val = 0.0f;
    if (m2 < SHT_N1) {
        int m  = (m2 < SHT_MMAX) ? m2 : (m2 - SHT_MMAX);
        int ph = (n * m) % SHT_K1;                 // exact: n*m < 2^24
        const float step  = 6.283185307179586f / 720.0f;
        const float scale = 6.283185307179586f / 720.0f;  // 2*pi * forward norm
        float ang = (float)ph * step;
        val = (m2 < SHT_MMAX) ? (cosf(ang) * scale) : (-sinf(ang) * scale);
    }
    Ft[idx] = val;
}

// ---------------------------------------------------------------------------
// Stage 1: Xf[plane][row] = X[row][:] . Ft[plane][:]   (K = 720, no ragged K)
// grid: (92416/128, 768/128=6), block: 128 threads = 4 waves
// ---------------------------------------------------------------------------
__global__ __launch_bounds__(128) void sht_dft_stage1(
        const float* __restrict__ X, const float* __restrict__ Ft,
        float* __restrict__ Xf) {
    __shared__ float As[2][BLK * LDSROW];
    __shared__ float Bs[2][BLK * LDSROW];

    const int tid  = threadIdx.x;
    const int lane = tid & 31;
    const int wave = tid >> 5;
    const int wr   = wave >> 1;          // wave row (0..1) -> 64-row strip
    const int wc   = wave & 1;           // wave col (0..1) -> 64-col strip
    const int mtile = blockIdx.x;        // 0..721
    const int ntile = blockIdx.y;        // 0..5

    const int mrow = lane & 15;
    const int koff = (lane >> 4) << 1;   // 0 or 2 (ISA 32-bit A/B fragment layout)

    v8f acc[4][4];
#pragma unroll
    for (int i = 0; i < 4; ++i)
#pragma unroll
        for (int j = 0; j < 4; ++j)
#pragma unroll
            for (int e = 0; e < 8; ++e) acc[i][j][e] = 0.0f;

    const float* Ag = X  + (size_t)(mtile * BLK) * SHT_K1 + (size_t)tid * SHT_K1;
    const float* Bg = Ft + (size_t)(ntile * BLK) * SHT_K1 + (size_t)tid * SHT_K1;

#ifdef SHT_HAVE_ASYNC_LDS
    // -------- async double-buffered pipeline: 8 b128 async loads / lane / tile
    {
        float* adst = &As[0][tid * LDSROW];
        float* bdst = &Bs[0][tid * LDSROW];
#pragma unroll
        for (int c = 0; c < 4; ++c) async_b128(Ag + c * 4, adst + c * 4);
#pragma unroll
        for (int c = 0; c < 4; ++c) async_b128(Bg + c * 4, bdst + c * 4);
    }
    int buf = 0;
    for (int kb = 0; kb < SHT_K1; kb += BLKK) {
        const int nkb = kb + BLKK;
        if (nkb < SHT_K1) {
            float* adst = &As[buf ^ 1][tid * LDSROW];
            float* bdst = &Bs[buf ^ 1][tid * LDSROW];
#pragma unroll
            for (int c = 0; c < 4; ++c) async_b128(Ag + nkb + c * 4, adst + c * 4);
#pragma unroll
            for (int c = 0; c < 4; ++c) async_b128(Bg + nkb + c * 4, bdst + c * 4);
            __builtin_amdgcn_s_wait_asynccnt(8);   // current tile's 8 loads done
        } else {
            __builtin_amdgcn_s_wait_asynccnt(0);
        }
        __syncthreads();                           // all waves' tile data visible

        const float* Asb = As[buf];
        const float* Bsb = Bs[buf];
#pragma unroll
        for (int kk = 0; kk < BLKK; kk += 4) {
            v2f af[4], bf[4];
#pragma unroll
            for (int i = 0; i < 4; ++i)
                af[i] = *(const v2f*)&Asb[(wr * 64 + i * 16 + mrow) * LDSROW + kk + koff];
#pragma unroll
            for (int j = 0; j < 4; ++j)
                bf[j] = *(const v2f*)&Bsb[(wc * 64 + j * 16 + mrow) * LDSROW + kk + koff];
#pragma unroll
            for (int i = 0; i < 4; ++i)
#pragma unroll
                for (int j = 0; j < 4; ++j)
                    acc[i][j] = __builtin_amdgcn_wmma_f32_16x16x4_f32(
                        false, af[i], false, bf[j], (short)0, acc[i][j], false, false);
        }
        __syncthreads();                           // tile fully consumed -> refill ok
        buf ^= 1;
    }
#else
    // -------- synchronous fallback (VGPR-staged global -> LDS)
    for (int kb = 0; kb < SHT_K1; kb += BLKK) {
        {
            const float* asrc = Ag + kb;
            float*       adst = &As[0][tid * LDSROW];
#pragma unroll
            for (int j = 0; j < BLKK; ++j) adst[j] = asrc[j];
            const float* bsrc = Bg + kb;
            float*       bdst = &Bs[0][tid * LDSROW];
#pragma unroll
            for (int j = 0; j < BLKK; ++j) bdst[j] = bsrc[j];
        }
        __syncthreads();
#pragma unroll
        for (int kk = 0; kk < BLKK; kk += 4) {
            v2f af[4], bf[4];
#pragma unroll
            for (int i = 0; i < 4; ++i)
                af[i] = *(const v2f*)&As[0][(wr * 64 + i * 16 + mrow) * LDSROW + kk + koff];
#pragma unroll
            for (int j = 0; j < 4; ++j)
                bf[j] = *(const v2f*)&Bs[0][(wc * 64 + j * 16 + mrow) * LDSROW + kk + koff];
#pragma unroll
            for (int i = 0; i < 4; ++i)
#pragma unroll
                for (int j = 0; j < 4; ++j)
                    acc[i][j] = __builtin_amdgcn_wmma_f32_16x16x4_f32(
                        false, af[i], false, bf[j], (short)0, acc[i][j], false, false);
        }
        __syncthreads();
    }
#endif

    // --- epilogue: Xf[plane][row], rows contiguous per lane -> float4 x2 ---
    const int rlo = (lane < 16) ? 0 : 8;
#pragma unroll
    for (int i = 0; i < 4; ++i) {
#pragma unroll
        for (int j = 0; j < 4; ++j) {
            int plane = ntile * BLK + wc * 64 + j * 16 + (lane & 15);
            if (plane < SHT_N1) {
                size_t addr = (size_t)plane * SHT_M1
                            + (size_t)(mtile * BLK + wr * 64 + i * 16 + rlo);
                v4f lo, hi;
#pragma unroll
                for (int e = 0; e < 4; ++e) { lo[e] = acc[i][j][e]; hi[e] = acc[i][j][e + 4]; }
                *(v4f*)&Xf[addr]     = lo;
                *(v4f*)&Xf[addr + 4] = hi;
            }
        }
    }
}

// ---------------------------------------------------------------------------
// Stage 2: per plane (re/im x m), out[c,l] = Xf[plane][c,:] . W[m][l,:]
// grid: (2 mtiles, 3 ntiles, 722 planes), block: 128 threads = 4 waves
// K = 361 (ragged: 22*16 + 9, zero-padded in LDS)
// ---------------------------------------------------------------------------
__global__ __launch_bounds__(128) void sht_leg_stage2(
        const float* __restrict__ Xf, const float* __restrict__ W,
        float* __restrict__ out) {
    __shared__ float As[BLK * LDSROW];
    __shared__ float Bs[BLK * LDSROW];

    const int tid  = threadIdx.x;
    const int lane = tid & 31;
    const int wave = tid >> 5;
    const int wr   = wave >> 1;
    const int wc   = wave & 1;
    const int mtile = blockIdx.x;        // 0..1  (C=256)
    const int ntile = blockIdx.y;        // 0..2  (L=360)
    const int plane = blockIdx.z;        // 0..721
    const int ri    = (plane >= SHT_MMAX) ? 1 : 0;
    const int m     = plane - ri * SHT_MMAX;
    const int bRowsValid = min(BLK, SHT_LMAX - ntile * BLK);

    const int mrow = lane & 15;
    const int koff = (lane >> 4) << 1;

    v8f acc[4][4];
#pragma unroll
    for (int i = 0; i < 4; ++i)
#pragma unroll
        for (int j = 0; j < 4; ++j)
#pragma unroll
            for (int e = 0; e < 8; ++e) acc[i][j][e] = 0.0f;

    const float* Ag = Xf + (size_t)plane * SHT_M1 + (size_t)(mtile * BLK) * SHT_NLAT;
    const float* Bg = W  + (size_t)m * SHT_LMAX * SHT_NLAT + (size_t)(ntile * BLK) * SHT_NLAT;

    for (int kb = 0; kb < SHT_NLAT; kb += BLKK) {
        const int kValid = min(BLKK, SHT_NLAT - kb);
        {
            const float* asrc = Ag + (size_t)tid * SHT_NLAT + kb;
            float*       adst = As + tid * LDSROW;
            __builtin_prefetch(asrc + BLKK, 0, 1);     // global_prefetch_b8 next chunk
#pragma unroll
            for (int j = 0; j < BLKK; ++j) adst[j] = (j < kValid) ? asrc[j] : 0.0f;

            const bool  brow = (tid < bRowsValid);
            const float* bsrc = Bg + (size_t)tid * SHT_NLAT + kb;
            float*       bdst = Bs + tid * LDSROW;
            __builtin_prefetch(bsrc + BLKK, 0, 1);
#pragma unroll
            for (int j = 0; j < BLKK; ++j) bdst[j] = (brow && j < kValid) ? bsrc[j] : 0.0f;
        }
        __syncthreads();

#pragma unroll
        for (int kk = 0; kk < BLKK; kk += 4) {
            v2f af[4], bf[4];
#pragma unroll
            for (int i = 0; i < 4; ++i)
                af[i] = *(const v2f*)&As[(wr * 64 + i * 16 + mrow) * LDSROW + kk + koff];
#pragma unroll
            for (int j = 0; j < 4; ++j)
                bf[j] = *(const v2f*)&Bs[(wc * 64 + j * 16 + mrow) * LDSROW + kk + koff];
#pragma unroll
            for (int i = 0; i < 4; ++i)
#pragma unroll
                for (int j = 0; j < 4; ++j)
                    acc[i][j] = __builtin_amdgcn_wmma_f32_16x16x4_f32(
                        false, af[i], false, bf[j], (short)0, acc[i][j], false, false);
        }
        __syncthreads();
    }

    // --- epilogue: scatter into interleaved complex out[((c*360+l)*361+m)*2+ri]
    const int rlo = (lane < 16) ? 0 : 8;
#pragma unroll
    for (int i = 0; i < 4; ++i) {
#pragma unroll
        for (int j = 0; j < 4; ++j) {
            int l = ntile * BLK + wc * 64 + j * 16 + (lane & 15);
            if (l < SHT_LMAX) {
                int c0 = mtile * BLK + wr * 64 + i * 16 + rlo;
#pragma unroll
                for (int e = 0; e < 8; ++e) {
                    size_t o = (((size_t)(c0 + e) * SHT_LMAX + l) * SHT_MMAX + m) * 2 + ri;
                    out[o] = acc[i][j][e];
                }
            }
        }
    }
}

// ---------------------------------------------------------------------------
extern "C" void kernel_launch(void* const* d_in, const int* in_sizes, int n_in,
                              void* d_out, int out_size, void* d_ws, size_t ws_size,
                              hipStream_t stream) {
    const float* X = (const float*)d_in[0];   // (1,256,361,720) f32
    const float* W = (const float*)d_in[1];   // (361,360,361)   f32
    float* out = (float*)d_out;               // (1,256,360,361) complex64 -> interleaved f32

    float* Ft = (float*)d_ws;                          // [768][720] (722 valid rows)
    float* Xf = Ft + (size_t)SHT_N1P * SHT_K1;         // [722][92416]

    {
        long total = (long)SHT_N1P * SHT_K1;
        int blocks = (int)((total + 255) / 256);
        sht_build_basis<<<blocks, 256, 0, stream>>>(Ft);
    }
    {
        dim3 grid(SHT_M1 / BLK, SHT_N1P / BLK);              // (722, 6)
        sht_dft_stage1<<<grid, 128, 0, stream>>>(X, Ft, Xf);
    }
    {
        dim3 grid(SHT_NC / BLK, (SHT_LMAX + BLK - 1) / BLK, SHT_N1);  // (2, 3, 722)
        sht_leg_stage2<<<grid, 128, 0, stream>>>(Xf, W, out);
    }
}